// ParticleFilterNetwork_68418829025980
// MI455X (gfx1250) — compile-verified
//
#include <hip/hip_runtime.h>
#include <hip/hip_bf16.h>
#include <math.h>

typedef __attribute__((ext_vector_type(2))) float v2f;
typedef __attribute__((ext_vector_type(8))) float v8f;

#define PF_N 512
#define PF_M 8192
#define PF_D 16
#define PF_THREADS 256
#define PF_NW (PF_THREADS / 32)
#define PF_NOISE 0.1f

__global__ __launch_bounds__(PF_THREADS) void pf_fused_kernel(
    const float* __restrict__ states_prev,   // [N,M,D]
    const float* __restrict__ lw_prev,       // [N,M]
    const float* __restrict__ observations,  // [N,D]
    const float* __restrict__ controls,      // [N,D]
    const float* __restrict__ noise,         // [N,M,D]
    const float* __restrict__ u,             // [N,M]
    const float* __restrict__ B_dyn,         // [D,D]
    const float* __restrict__ W_meas,        // [D,D]
    float* __restrict__ out_est,             // [N,D]
    float* __restrict__ out_states,          // [N,M,D]
    float* __restrict__ out_lw)              // [N,M]
{
    __shared__ float s_buf[PF_M];      // lw, then CDF in place (32 KB; WGP has 320 KB)
    __shared__ float s_cB[PF_D];
    __shared__ float s_obs[PF_D];
    __shared__ float s_est[PF_D];
    __shared__ float s_redmax[PF_NW];
    __shared__ float s_redsum[PF_NW];
    __shared__ float s_wtot[PF_NW];

    const int n    = blockIdx.x;
    const int tid  = threadIdx.x;
    const int wv   = tid >> 5;
    const int lane = tid & 31;
    const int hi   = lane >> 4;   // which 16-lane half of the wave
    const int lp   = lane & 15;
    const size_t rowM = (size_t)n * PF_M;
    const size_t rowP = rowM * PF_D;

    // ---- phase 0: per-row constants -------------------------------------
    if (tid < PF_D) {
        float acc = 0.f;
        #pragma unroll
        for (int k = 0; k < PF_D; ++k)
            acc += controls[n * PF_D + k] * B_dyn[k * PF_D + tid];
        s_cB[tid]  = acc;
        s_obs[tid] = observations[n * PF_D + tid];
        s_est[tid] = 0.f;
    }
    __syncthreads();

    // WMMA computes pred_obs TRANSPOSED:  D(16x16) = W^T(16xK) x sp^T(Kx16).
    //   A-fragment (16x4): lane holds W[4c+2*hi + j][lp]      (= W^T rows)
    //   B-fragment (4x16): lane holds sp[base+lp][4c+2*hi + j] (= sp^T cols)
    //   C/D: lane's acc[v] = pred_obs[particle base+lp][dim v + 8*hi]
    // => each lane owns all dims of ITS OWN particle; the ||err||^2 reduction
    //    is 8 in-lane FMAs + one xor-16 half-swap (no multi-step ds_bpermute).
    v2f afrW[4], cbv[4];
    #pragma unroll
    for (int c = 0; c < 4; ++c) {
        const int d0 = 4 * c + 2 * hi;
        afrW[c][0] = W_meas[(d0 + 0) * PF_D + lp];
        afrW[c][1] = W_meas[(d0 + 1) * PF_D + lp];
        cbv[c][0]  = s_cB[d0 + 0];
        cbv[c][1]  = s_cB[d0 + 1];
    }
    float obs8[8];
    #pragma unroll
    for (int v = 0; v < 8; ++v) obs8[v] = s_obs[v + 8 * hi];

    // ---- phase 1: streaming pass: states_pred -> WMMA pred_obs -> lw ----
    // Online logsumexp + online weighted-state accumulation (one HBM pass).
    float m_run = -__builtin_inff();
    float s_run = 0.f;
    float accd[8];
    #pragma unroll
    for (int k = 0; k < 8; ++k) accd[k] = 0.f;

    #pragma unroll 1
    for (int t = 0; t < PF_M / (16 * PF_NW); ++t) {   // 64 iterations
        const int base = (t * PF_NW + wv) * 16;       // this wave's 16-particle tile
        const int p    = base + lp;
        const size_t poff = rowP + (size_t)p * PF_D;

        // Build sp^T B-fragments of states_pred directly in WMMA layout.
        v2f b[4];
        #pragma unroll
        for (int c = 0; c < 4; ++c) {
            const int d0 = 4 * c + 2 * hi;
            const float2 pv = *(const float2*)(states_prev + poff + d0);
            const float2 nv = *(const float2*)(noise       + poff + d0);
            b[c][0] = pv.x + cbv[c][0] + PF_NOISE * nv.x;
            b[c][1] = pv.y + cbv[c][1] + PF_NOISE * nv.y;
        }

        // coalesced (broadcast across halves) previous log-weight
        const float lwp = lw_prev[rowM + p];

        // pred_obs^T = W^T @ sp^T, K in 4 chunks of 4
        v8f acc = {0.f, 0.f, 0.f, 0.f, 0.f, 0.f, 0.f, 0.f};
        #pragma unroll
        for (int c = 0; c < 4; ++c)
            acc = __builtin_amdgcn_wmma_f32_16x16x4_f32(
                false, afrW[c], false, b[c], (short)0, acc, false, false);

        // ||pred_obs[p] - obs||^2 : 8 dims in-lane, other 8 from lane^16
        float ss = 0.f;
        #pragma unroll
        for (int v = 0; v < 8; ++v) {
            const float d = acc[v] - obs8[v];
            ss = fmaf(d, d, ss);
        }
        ss += __shfl_xor(ss, 16, 32);

        const float lw = lwp - 0.5f * ss;
        if (hi == 0) s_buf[p] = lw;    // 16 lanes store the tile's 16 lw values

        // online logsumexp + weighted state (lane holds dims 4c+2*hi,+1 of p)
        const float nm = fmaxf(m_run, lw);
        const float sc = __expf(m_run - nm);
        const float e  = __expf(lw - nm);
        s_run = s_run * sc + e;
        #pragma unroll
        for (int c = 0; c < 4; ++c) {
            accd[2 * c + 0] = accd[2 * c + 0] * sc + e * b[c][0];
            accd[2 * c + 1] = accd[2 * c + 1] * sc + e * b[c][1];
        }
        m_run = nm;
    }
    __syncthreads();

    // ---- phase 2: block logsumexp + state estimate ----------------------
    float wm = m_run;
    wm = fmaxf(wm, __shfl_xor(wm, 1, 32));
    wm = fmaxf(wm, __shfl_xor(wm, 2, 32));
    wm = fmaxf(wm, __shfl_xor(wm, 4, 32));
    wm = fmaxf(wm, __shfl_xor(wm, 8, 32));
    wm = fmaxf(wm, __shfl_xor(wm, 16, 32));
    if (lane == 0) s_redmax[wv] = wm;
    __syncthreads();
    float Mb = s_redmax[0];
    #pragma unroll
    for (int k = 1; k < PF_NW; ++k) Mb = fmaxf(Mb, s_redmax[k]);

    const float sc2 = __expf(m_run - Mb);
    // each particle is owned by two lanes (l, l+16); count Z only from hi==0
    float sz = (hi == 0) ? s_run * sc2 : 0.f;
    sz += __shfl_xor(sz, 1, 32);
    sz += __shfl_xor(sz, 2, 32);
    sz += __shfl_xor(sz, 4, 32);
    sz += __shfl_xor(sz, 8, 32);
    sz += __shfl_xor(sz, 16, 32);
    if (lane == 0) s_redsum[wv] = sz;
    #pragma unroll
    for (int k = 0; k < 8; ++k) {
        const int dim = 4 * (k >> 1) + 2 * hi + (k & 1);
        atomicAdd(&s_est[dim], accd[k] * sc2);   // ds_add_f32
    }
    __syncthreads();
    float Z = 0.f;
    #pragma unroll
    for (int k = 0; k < PF_NW; ++k) Z += s_redsum[k];
    const float LSE = Mb + __logf(Z);
    if (tid < PF_D) out_est[n * PF_D + tid] = s_est[tid] / Z;

    // ---- phase 3: in-LDS CDF (chunked block-wide inclusive scan) --------
    float carry = 0.f;
    #pragma unroll 1
    for (int ch = 0; ch < PF_M / PF_THREADS; ++ch) {   // 32 chunks
        const int i = ch * PF_THREADS + tid;
        float scn = __expf(s_buf[i] - LSE);
        #pragma unroll
        for (int off = 1; off < 32; off <<= 1) {
            float tv = __shfl_up(scn, off, 32);
            if (lane >= off) scn += tv;
        }
        if (lane == 31) s_wtot[wv] = scn;
        __syncthreads();
        float pre = 0.f, tot = 0.f;
        #pragma unroll
        for (int k = 0; k < PF_NW; ++k) {
            const float v = s_wtot[k];
            tot += v;
            if (k < wv) pre += v;
        }
        s_buf[i] = carry + pre + scn;
        carry += tot;
        __syncthreads();
    }

    // ---- phase 4: inverse-CDF resample + gather-recompute ---------------
    const float lwc = -9.0109131f;   // -log(8192)
    #pragma unroll 1
    for (int m0 = tid; m0 < PF_M; m0 += PF_THREADS) {
        const float um = u[rowM + m0];
        int lo = 0, hsz = PF_M;
        #pragma unroll 1
        while (lo < hsz) {           // 13 or 14 steps over LDS CDF
            const int mid = (lo + hsz) >> 1;
            if (s_buf[mid] < um) lo = mid + 1; else hsz = mid;
        }
        const int idx = (lo < PF_M) ? lo : (PF_M - 1);
        const size_t g = rowP + (size_t)idx * PF_D;
        const size_t o = rowP + (size_t)m0  * PF_D;
        #pragma unroll
        for (int k = 0; k < 4; ++k) {
            const float4 pv = *(const float4*)(states_prev + g + 4 * k);
            const float4 nv = *(const float4*)(noise       + g + 4 * k);
            float4 r;
            r.x = pv.x + s_cB[4 * k + 0] + PF_NOISE * nv.x;
            r.y = pv.y + s_cB[4 * k + 1] + PF_NOISE * nv.y;
            r.z = pv.z + s_cB[4 * k + 2] + PF_NOISE * nv.z;
            r.w = pv.w + s_cB[4 * k + 3] + PF_NOISE * nv.w;
            *(float4*)(out_states + o + 4 * k) = r;
        }
        out_lw[rowM + m0] = lwc;
    }
}

extern "C" void kernel_launch(void* const* d_in, const int* in_sizes, int n_in,
                              void* d_out, int out_size, void* d_ws, size_t ws_size,
                              hipStream_t stream) {
    (void)in_sizes; (void)n_in; (void)out_size; (void)d_ws; (void)ws_size;
    const float* states_prev  = (const float*)d_in[0];
    const float* lw_prev      = (const float*)d_in[1];
    const float* observations = (const float*)d_in[2];
    const float* controls     = (const float*)d_in[3];
    const float* noise        = (const float*)d_in[4];
    const float* u            = (const float*)d_in[5];
    const float* B_dyn        = (const float*)d_in[6];
    const float* W_meas       = (const float*)d_in[7];

    float* out        = (float*)d_out;
    float* out_est    = out;                                     // [N,D]
    float* out_states = out + (size_t)PF_N * PF_D;               // [N,M,D]
    float* out_lw     = out_states + (size_t)PF_N * PF_M * PF_D; // [N,M]

    hipLaunchKernelGGL(pf_fused_kernel, dim3(PF_N), dim3(PF_THREADS), 0, stream,
                       states_prev, lw_prev, observations, controls, noise, u,
                       B_dyn, W_meas, out_est, out_states, out_lw);
}